// LinksPredictor_25005299597337
// MI455X (gfx1250) — compile-verified
//
#include <hip/hip_runtime.h>

typedef __attribute__((ext_vector_type(2))) float v2f;
typedef __attribute__((ext_vector_type(4))) float v4f;
typedef __attribute__((ext_vector_type(8))) float v8f;

#define HIDDEN  128
#define N_NODES 100000
#define N_EDGES 500000

// ---------------------------------------------------------------------------
// Kernel 1: per-node projection  Y[m][n] = sum_k X[m][k] * W[n][k] + b[n]
// (i.e. Y = X @ W^T + b), done with v_wmma_f32_16x16x4_f32.
//
// Block = 256 threads = 8 waves. Block -> one 16-row M tile; wave w -> N tile
// [16w, 16w+16). K=128 accumulated as 32 chained 16x16x4 fp32 WMMAs.
//
// A fragment (16x4 f32): lane l<16 holds X[m0+l][k+0..1], lane l+16 holds
// X[m0+l][k+2..3]  -> per-lane float2 load from row-major X.
// B fragment (4x16): lane l<16 holds W[n0+l][k+0..1] etc. -> identical
// per-lane float2 pattern from row-major W (GEMM is against W^T).
// C/D (16x16 f32): VGPR r, lane l: row m0 + r + 8*(l>=16), col n0 + (l&15).
// ---------------------------------------------------------------------------
__global__ __launch_bounds__(256)
void proj_wmma_kernel(const float* __restrict__ X,
                      const float* __restrict__ W,
                      const float* __restrict__ bias,
                      float* __restrict__ Y)
{
    const int wave = threadIdx.x >> 5;   // 0..7 -> N tile
    const int lane = threadIdx.x & 31;
    const int half = lane >> 4;          // 0: K pair {0,1}, 1: K pair {2,3}
    const int l15  = lane & 15;

    const int m0 = blockIdx.x * 16;
    const int n0 = wave * 16;

    const float* xrow = X + (size_t)(m0 + l15) * HIDDEN + 2 * half;
    const float* wrow = W + (size_t)(n0 + l15) * HIDDEN + 2 * half;

    v8f acc = {};
#pragma unroll
    for (int k = 0; k < HIDDEN; k += 4) {
        v2f a = *(const v2f*)(xrow + k);
        v2f b = *(const v2f*)(wrow + k);
        // 8 args: (neg_a, A, neg_b, B, c_mod, C, reuse_a, reuse_b)
        acc = __builtin_amdgcn_wmma_f32_16x16x4_f32(
            false, a, false, b, (short)0, acc, false, false);
    }

    const float bv = bias[n0 + l15];

    // Store: VGPR r maps to row m0 + r + 8*half, column n0 + l15.
    float* yout = Y + (size_t)(m0 + 8 * half) * HIDDEN + n0 + l15;
#pragma unroll
    for (int r = 0; r < 8; ++r) {
        yout[(size_t)r * HIDDEN] = acc[r] + bv;
    }
}

// ---------------------------------------------------------------------------
// Kernel 2: per-edge gather + dot.  out[e] = dot(qa[src[e]], qb[dst[e]])
// One wave32 per edge: lane i handles 4 floats (float4), then 5-step
// shfl_xor reduction. qa/qb working set (102 MB) is L2-resident (192 MB L2),
// so the random gathers replay out of L2, not HBM.
// ---------------------------------------------------------------------------
__global__ __launch_bounds__(256)
void edge_dot_kernel(const long long* __restrict__ idx,   // [2, N_EDGES] int64
                     const float* __restrict__ qa,
                     const float* __restrict__ qb,
                     float* __restrict__ out)
{
    const int wave = threadIdx.x >> 5;
    const int lane = threadIdx.x & 31;
    const int e    = blockIdx.x * 8 + wave;

    const long long s = idx[e];
    const long long d = idx[(size_t)N_EDGES + e];

    const v4f a = *(const v4f*)(qa + (size_t)s * HIDDEN + lane * 4);
    const v4f b = *(const v4f*)(qb + (size_t)d * HIDDEN + lane * 4);

    float sum = a.x * b.x + a.y * b.y + a.z * b.z + a.w * b.w;

#pragma unroll
    for (int off = 16; off > 0; off >>= 1)
        sum += __shfl_xor(sum, off, 32);

    if (lane == 0) out[e] = sum;
}

// ---------------------------------------------------------------------------
// Launch: project A-side and B-side nodes into workspace, then edge dot.
// ws layout: qa = ws[0 .. 51.2MB), qb = ws[51.2MB .. 102.4MB).
// Same-stream ordering serializes the GEMMs before the edge kernel.
// ---------------------------------------------------------------------------
extern "C" void kernel_launch(void* const* d_in, const int* in_sizes, int n_in,
                              void* d_out, int out_size, void* d_ws, size_t ws_size,
                              hipStream_t stream)
{
    const float*     nfa  = (const float*)d_in[0];      // [N_NODES, HIDDEN]
    const float*     nfb  = (const float*)d_in[1];      // [N_NODES, HIDDEN]
    const long long* eidx = (const long long*)d_in[2];  // [2, N_EDGES] int64
    const float*     Wa   = (const float*)d_in[3];      // [HIDDEN, HIDDEN]
    const float*     ba   = (const float*)d_in[4];      // [HIDDEN]
    const float*     Wb   = (const float*)d_in[5];      // [HIDDEN, HIDDEN]
    const float*     bb   = (const float*)d_in[6];      // [HIDDEN]
    float*           out  = (float*)d_out;              // [N_EDGES]

    float* qa = (float*)d_ws;
    float* qb = qa + (size_t)N_NODES * HIDDEN;

    dim3 blk(256);
    proj_wmma_kernel<<<N_NODES / 16, blk, 0, stream>>>(nfa, Wa, ba, qa);
    proj_wmma_kernel<<<N_NODES / 16, blk, 0, stream>>>(nfb, Wb, bb, qb);
    edge_dot_kernel<<<N_EDGES / 8, blk, 0, stream>>>(eidx, qa, qb, out);

    (void)in_sizes; (void)n_in; (void)out_size; (void)ws_size;
}